// Cond_PlanarTrans_8572754722980
// MI455X (gfx1250) — compile-verified
//
#include <hip/hip_runtime.h>

// Cond planar transform, O_DIM==1, N_M==8.
// out = s + u * tanh(w*s + b) where (w,u,b) = relu(o*W[m] + bias[m]) from an
// 8-entry parameter table. Pure streaming elementwise op: 16 B/elem moved,
// HBM-bound (~5.8 us floor at 23.3 TB/s for 8.39M elems).

typedef float vf4 __attribute__((ext_vector_type(4)));
typedef int   vi4 __attribute__((ext_vector_type(4)));
typedef float vf2 __attribute__((ext_vector_type(2)));

#define DEVFN __device__ __forceinline__

DEVFN float fast_tanh(float x) {
#if __has_builtin(__builtin_amdgcn_tanhf)
  return __builtin_amdgcn_tanhf(x);   // gfx1250 native v_tanh_f32
#else
  return tanhf(x);
#endif
}

// Per-element math. tbl points at LDS: entry n = {W1,b1,W2,b2,W3,b3} (24 B).
DEVFN float planar_elem(int idx, float s, float o, const float* tbl) {
  const vf2* rec = (const vf2*)(tbl + idx * 6);  // 8-byte aligned (24B stride)
  vf2 wb1 = rec[0];   // ds_load_b64
  vf2 wb2 = rec[1];
  vf2 wb3 = rec[2];
  float w = fmaxf(__builtin_fmaf(o, wb1.x, wb1.y), 0.0f);
  float u = fmaxf(__builtin_fmaf(o, wb2.x, wb2.y), 0.0f);
  float b = fmaxf(__builtin_fmaf(o, wb3.x, wb3.y), 0.0f);
  float t = fast_tanh(__builtin_fmaf(w, s, b));
  return __builtin_fmaf(u, t, s);
}

__global__ void __launch_bounds__(256)
cond_planar_vec4(const int*   __restrict__ m,
                 const float* __restrict__ s,
                 const float* __restrict__ o,
                 const float* __restrict__ W1, const float* __restrict__ b1,
                 const float* __restrict__ W2, const float* __restrict__ b2,
                 const float* __restrict__ W3, const float* __restrict__ b3,
                 float* __restrict__ out, int nvec) {
  __shared__ float tbl[8 * 6];
  const int t = threadIdx.x;
  if (t < 8) {  // fill 8-entry packed parameter table (48 floats)
    tbl[t * 6 + 0] = W1[t];
    tbl[t * 6 + 1] = b1[t];
    tbl[t * 6 + 2] = W2[t];
    tbl[t * 6 + 3] = b2[t];
    tbl[t * 6 + 4] = W3[t];
    tbl[t * 6 + 5] = b3[t];
  }
  __syncthreads();

  const int i = blockIdx.x * 256 + t;
  if (i >= nvec) return;

  // 128-bit non-temporal streaming loads: 3 x b128 in, 1 x b128 out.
  vi4 mi = __builtin_nontemporal_load((const vi4*)m + i);
  vf4 sv = __builtin_nontemporal_load((const vf4*)s + i);
  vf4 ov = __builtin_nontemporal_load((const vf4*)o + i);

  vf4 r;
  r.x = planar_elem(mi.x, sv.x, ov.x, tbl);
  r.y = planar_elem(mi.y, sv.y, ov.y, tbl);
  r.z = planar_elem(mi.z, sv.z, ov.z, tbl);
  r.w = planar_elem(mi.w, sv.w, ov.w, tbl);

  __builtin_nontemporal_store(r, (vf4*)out + i);
}

// Tail kernel for N % 4 != 0 (not hit for B=128,P=65536 but kept for safety).
__global__ void __launch_bounds__(64)
cond_planar_tail(const int*   __restrict__ m,
                 const float* __restrict__ s,
                 const float* __restrict__ o,
                 const float* __restrict__ W1, const float* __restrict__ b1,
                 const float* __restrict__ W2, const float* __restrict__ b2,
                 const float* __restrict__ W3, const float* __restrict__ b3,
                 float* __restrict__ out, int start, int n) {
  int i = start + blockIdx.x * 64 + threadIdx.x;
  if (i >= n) return;
  int   idx = m[i];
  float sv  = s[i];
  float ov  = o[i];
  float w = fmaxf(__builtin_fmaf(ov, W1[idx], b1[idx]), 0.0f);
  float u = fmaxf(__builtin_fmaf(ov, W2[idx], b2[idx]), 0.0f);
  float b = fmaxf(__builtin_fmaf(ov, W3[idx], b3[idx]), 0.0f);
  out[i] = __builtin_fmaf(u, fast_tanh(__builtin_fmaf(w, sv, b)), sv);
}

extern "C" void kernel_launch(void* const* d_in, const int* in_sizes, int n_in,
                              void* d_out, int out_size, void* d_ws, size_t ws_size,
                              hipStream_t stream) {
  const int*   m  = (const int*)  d_in[0];
  const float* s  = (const float*)d_in[1];
  const float* o  = (const float*)d_in[2];
  const float* W1 = (const float*)d_in[3];
  const float* b1 = (const float*)d_in[4];
  const float* W2 = (const float*)d_in[5];
  const float* b2 = (const float*)d_in[6];
  const float* W3 = (const float*)d_in[7];
  const float* b3 = (const float*)d_in[8];
  float* out = (float*)d_out;

  const int n    = in_sizes[1];     // B*P*O_DIM = 8,388,608
  const int nvec = n / 4;
  const int rem  = n - nvec * 4;

  if (nvec > 0) {
    dim3 grid((nvec + 255) / 256);
    cond_planar_vec4<<<grid, 256, 0, stream>>>(m, s, o, W1, b1, W2, b2, W3, b3,
                                               out, nvec);
  }
  if (rem > 0) {
    cond_planar_tail<<<1, 64, 0, stream>>>(m, s, o, W1, b1, W2, b2, W3, b3,
                                           out, nvec * 4, n);
  }
}